// TopK_37125697306644
// MI455X (gfx1250) — compile-verified
//
#include <hip/hip_runtime.h>
#include <stdint.h>

// Row-wise top-k keep: out[r, i] = x[r, i] if x[r, i] is among the K largest
// in row r (ties by lowest index), else 0.  Rows of 4096 fp32, K = 512.
//
// MI455X plan: pure bandwidth problem (512 MB traffic @ 23.3 TB/s ~ 22 us).
// Use CDNA5 async global<->LDS DMA path for perfectly coalesced staging,
// LDS-atomic radix select (4 x 8-bit digits) for the exact threshold,
// exact index-ordered tie-break.

#define ROWLEN    4096
#define KEEP      512
#define NTHREADS  256
#define PER_TH    (ROWLEN / NTHREADS)   // 16 elements per thread

__device__ __forceinline__ uint32_t ord_key(uint32_t b) {
    // order-preserving map: float bits -> unsigned (larger float => larger key)
    return b ^ ((b & 0x80000000u) ? 0xFFFFFFFFu : 0x80000000u);
}
__device__ __forceinline__ uint32_t inv_key(uint32_t u) {
    return u ^ ((u & 0x80000000u) ? 0x80000000u : 0xFFFFFFFFu);
}

__global__ __launch_bounds__(NTHREADS)
void topk_keep_rows(const float* __restrict__ x, float* __restrict__ out, int nrows)
{
    __shared__ __align__(16) float sh_data[ROWLEN];   // 16 KB staging buffer
    __shared__ unsigned sh_hist[256];
    __shared__ unsigned sh_g16[16];
    __shared__ unsigned sh_cnt[NTHREADS];
    __shared__ unsigned sh_prefix;
    __shared__ unsigned sh_krem;

    const int row = (int)blockIdx.x;
    if (row >= nrows) return;                         // uniform per block

    const int tid  = (int)threadIdx.x;
    const int wave = tid >> 5;                        // wave32
    const int lane = tid & 31;

    const unsigned long long srcRow =
        (unsigned long long)(uintptr_t)(x + (size_t)row * ROWLEN);
    const unsigned long long dstRow =
        (unsigned long long)(uintptr_t)(out + (size_t)row * ROWLEN);
    const uint32_t ldsBase = (uint32_t)(uintptr_t)&sh_data[0];

    // ---- 1) async DMA: global -> LDS, 16B/lane, contiguous 512B per wave op
    #pragma unroll
    for (int c = 0; c < 4; ++c) {
        uint32_t off   = (uint32_t)(wave * 2048 + c * 512 + lane * 16);
        uint32_t laddr = ldsBase + off;
        asm volatile("global_load_async_to_lds_b128 %0, %1, %2"
                     :: "v"(laddr), "v"(off), "s"(srcRow) : "memory");
    }
    asm volatile("s_wait_asynccnt 0" ::: "memory");
    __syncthreads();

    // ---- 2) my 16 contiguous elements -> registers as order keys
    uint32_t u[PER_TH];
    {
        const uint4* p = (const uint4*)&sh_data[tid * PER_TH];
        #pragma unroll
        for (int c = 0; c < 4; ++c) {
            uint4 q = p[c];
            u[c*4+0] = ord_key(q.x);
            u[c*4+1] = ord_key(q.y);
            u[c*4+2] = ord_key(q.z);
            u[c*4+3] = ord_key(q.w);
        }
    }

    if (tid == 0) { sh_prefix = 0u; sh_krem = (unsigned)KEEP; }

    // ---- 3) radix select, MSB-first, 4 passes of 8 bits
    #pragma unroll 1
    for (int d = 3; d >= 0; --d) {
        sh_hist[tid] = 0u;
        __syncthreads();
        const unsigned pfx   = sh_prefix;
        const int      shift = d * 8;
        #pragma unroll
        for (int j = 0; j < PER_TH; ++j) {
            bool match = (d == 3) || (((u[j] ^ pfx) >> (shift + 8)) == 0u);
            if (match) atomicAdd(&sh_hist[(u[j] >> shift) & 0xFFu], 1u);
        }
        __syncthreads();
        if (tid < 16) {                                // 16-bin group sums
            unsigned s = 0;
            #pragma unroll
            for (int i = 0; i < 16; ++i) s += sh_hist[tid * 16 + i];
            sh_g16[tid] = s;
        }
        __syncthreads();
        if (tid == 0) {                                // ~32 serial steps
            unsigned kr = sh_krem;
            int g = 15;
            while (sh_g16[g] < kr) { kr -= sh_g16[g]; --g; }
            int b = g * 16 + 15;
            while (sh_hist[b] < kr) { kr -= sh_hist[b]; --b; }
            sh_prefix = pfx | (((unsigned)b) << shift);
            sh_krem   = kr;
        }
        __syncthreads();
    }

    const unsigned thresh = sh_prefix;  // key of the KEEP-th largest element
    const unsigned krem   = sh_krem;    // # of threshold-equal elems to keep

    // ---- 4) exact tie-break by index: exclusive prefix of equal-counts
    unsigned cnt = 0;
    #pragma unroll
    for (int j = 0; j < PER_TH; ++j) cnt += (u[j] == thresh) ? 1u : 0u;
    sh_cnt[tid] = cnt;
    __syncthreads();
    unsigned base = 0;
    if (cnt) {                                         // rare: usually 1 thread
        for (int t = 0; t < tid; ++t) base += sh_cnt[t];
    }

    // ---- 5) produce output into my own LDS slice
    {
        float4* p = (float4*)&sh_data[tid * PER_TH];
        unsigned r = base;
        #pragma unroll
        for (int c = 0; c < 4; ++c) {
            float o[4];
            #pragma unroll
            for (int e = 0; e < 4; ++e) {
                unsigned uv = u[c*4 + e];
                bool keep;
                if (uv > thresh)       keep = true;
                else if (uv == thresh) { keep = (r < krem); ++r; }
                else                   keep = false;
                o[e] = keep ? __uint_as_float(inv_key(uv)) : 0.0f;
            }
            p[c] = make_float4(o[0], o[1], o[2], o[3]);
        }
    }

    // ---- 6) async DMA: LDS -> global (my wave's slice only; no barrier needed)
    asm volatile("s_wait_dscnt 0" ::: "memory");
    #pragma unroll
    for (int c = 0; c < 4; ++c) {
        uint32_t off   = (uint32_t)(wave * 2048 + c * 512 + lane * 16);
        uint32_t laddr = ldsBase + off;
        asm volatile("global_store_async_from_lds_b128 %0, %1, %2"
                     :: "v"(off), "v"(laddr), "s"(dstRow) : "memory");
    }
    asm volatile("s_wait_asynccnt 0" ::: "memory");
}

extern "C" void kernel_launch(void* const* d_in, const int* in_sizes, int n_in,
                              void* d_out, int out_size, void* d_ws, size_t ws_size,
                              hipStream_t stream) {
    const float* x  = (const float*)d_in[0];
    float* out      = (float*)d_out;
    const int nrows = in_sizes[0] / ROWLEN;            // 16384 for (4,4096,4096)
    topk_keep_rows<<<nrows, NTHREADS, 0, stream>>>(x, out, nrows);
}